// FourierMixer_24318104830063
// MI455X (gfx1250) — compile-verified
//
#include <hip/hip_runtime.h>
#include <hip/hip_bf16.h>

// FourierMixer: ifft2(fft2(x)).real on real float32 input == identity.
// Optimal MI455X implementation is a full-bandwidth streaming copy
// (268 MB total @ 23.3 TB/s ~= 11.5 us). We route it through the CDNA5
// async global<->LDS data mover (ASYNCcnt path) with 128b-per-lane ops.

typedef int v4i __attribute__((ext_vector_type(4)));
typedef __attribute__((address_space(1))) v4i gv4i;   // global int4
typedef __attribute__((address_space(3))) v4i lv4i;   // LDS int4

#if __has_builtin(__builtin_amdgcn_global_load_async_to_lds_b128) && \
    __has_builtin(__builtin_amdgcn_global_store_async_from_lds_b128)
#define USE_ASYNC_LDS 1
#else
#define USE_ASYNC_LDS 0
#endif

#if __has_builtin(__builtin_amdgcn_s_wait_asynccnt)
#define WAIT_ASYNC0() __builtin_amdgcn_s_wait_asynccnt(0)
#else
#define WAIT_ASYNC0() ((void)0)
#endif

__global__ __launch_bounds__(256) void fouriermixer_identity_copy(
    const float4* __restrict__ src, float4* __restrict__ dst,
    unsigned long long nvec)
{
#if USE_ASYNC_LDS
    // One float4 slot per thread, double-buffered: 2 * 256 * 16B = 8 KB LDS.
    __shared__ float4 stage[2][256];
    const unsigned t = threadIdx.x;
    lv4i* l0 = (lv4i*)&stage[0][t];
    lv4i* l1 = (lv4i*)&stage[1][t];

    const unsigned long long stride =
        (unsigned long long)gridDim.x * (unsigned long long)blockDim.x;
    unsigned long long i =
        (unsigned long long)blockIdx.x * (unsigned long long)blockDim.x + t;

    // Prologue: start load of tile 0 into buffer 0.
    if (i < nvec) {
        __builtin_amdgcn_global_load_async_to_lds_b128(
            (gv4i*)(src + i), l0, /*offset=*/0, /*cpol=*/0);
    }

    int cur = 0;
    while (i < nvec) {
        const unsigned long long nexti = i + stride;
        // Drain ASYNCcnt: current tile's load is complete (and any prior
        // store that read this buffer has finished), so LDS slot is ready.
        WAIT_ASYNC0();
        lv4i* lcur = cur ? l1 : l0;
        lv4i* lnxt = cur ? l0 : l1;
        // Kick the store of the current tile...
        __builtin_amdgcn_global_store_async_from_lds_b128(
            (gv4i*)(dst + i), lcur, /*offset=*/0, /*cpol=*/0);
        // ...and overlap it with the load of the next tile into the other buffer.
        if (nexti < nvec) {
            __builtin_amdgcn_global_load_async_to_lds_b128(
                (gv4i*)(src + nexti), lnxt, /*offset=*/0, /*cpol=*/0);
        }
        cur ^= 1;
        i = nexti;
    }
    // Make sure all async stores have left the wave before it retires.
    WAIT_ASYNC0();
#else
    // Fallback (host pass / toolchains without the async builtins):
    // plain wide streaming copy, still HBM-bound.
    const unsigned long long stride =
        (unsigned long long)gridDim.x * (unsigned long long)blockDim.x;
    for (unsigned long long i =
             (unsigned long long)blockIdx.x * blockDim.x + threadIdx.x;
         i < nvec; i += stride) {
        dst[i] = src[i];
    }
#endif
}

extern "C" void kernel_launch(void* const* d_in, const int* in_sizes, int n_in,
                              void* d_out, int out_size, void* d_ws, size_t ws_size,
                              hipStream_t stream) {
    const float4* src = (const float4*)d_in[0];
    float4* dst = (float4*)d_out;

    const unsigned long long n    = (unsigned long long)in_sizes[0]; // 8*4096*1024, divisible by 4
    const unsigned long long nvec = n >> 2;                          // float4 count

    const int block = 256;                      // 8 wave32s per block
    const unsigned long long per_thread = 8;    // 8 x float4 = 128 B per thread
    unsigned long long blocks =
        (nvec + (unsigned long long)block * per_thread - 1) /
        ((unsigned long long)block * per_thread);
    if (blocks < 1) blocks = 1;

    fouriermixer_identity_copy<<<dim3((unsigned)blocks), dim3(block), 0, stream>>>(
        src, dst, nvec);
}